// QuantizeEMA_90787018703329
// MI455X (gfx1250) — compile-verified
//
#include <hip/hip_runtime.h>
#include <hip/hip_bf16.h>

typedef __attribute__((ext_vector_type(2))) float v2f;
typedef __attribute__((ext_vector_type(8))) float v8f;

#define VQ_DIM      64
#define VQ_NEMBED   8192
#define VQ_NPIX     16384          // 16 * 32 * 32
#define VQ_HW       1024           // 32*32
#define VQ_IMGSTRIDE (VQ_DIM * VQ_HW)

// ---------------------------------------------------------------------------
// Kernel 1: precompute ||e_j||^2 for all 8192 codes; zero the diff slot.
// ---------------------------------------------------------------------------
__global__ __launch_bounds__(256) void vq_prep_kernel(
    const float* __restrict__ emb, float* __restrict__ e2,
    float* __restrict__ diff_slot) {
  int j = blockIdx.x * 256 + threadIdx.x;
  if (j < VQ_NEMBED) {
    float s = 0.0f;
#pragma unroll
    for (int c = 0; c < VQ_DIM; ++c) {
      float e = emb[c * VQ_NEMBED + j];
      s += e * e;
    }
    e2[j] = s;
  }
  if (blockIdx.x == 0 && threadIdx.x == 0) *diff_slot = 0.0f;
}

// ---------------------------------------------------------------------------
// Kernel 2: distance GEMM via V_WMMA_F32_16X16X4_F32 + in-register argmin.
// Each wave owns TWO 16-row M-tiles (A register-resident) and scans all 8192
// codes in 16-column chunks, sharing each B fragment across both tiles.
// Grid: 64 blocks x 256 threads (8 waves) -> 512 waves -> 1024 M-tiles.
// ---------------------------------------------------------------------------
__global__ __launch_bounds__(256) void vq_argmin_kernel(
    const float* __restrict__ inp,   // [16, 64, 32, 32]
    const float* __restrict__ emb,   // [64, 8192]
    const float* __restrict__ e2,    // [8192]
    int* __restrict__ ind) {         // [16384]
  const int lane  = threadIdx.x & 31;
  const int wave  = threadIdx.x >> 5;
  const int gw    = blockIdx.x * 8 + wave;   // 0..511
  const int tile0 = gw * 2;                  // two consecutive M-tiles
  const int mrow  = lane & 15;               // A-matrix M for this lane
  const int khalf = (lane >> 4) * 2;         // lanes 16..31 hold K+2,K+3

  // --- Load A fragments for both tiles, register resident for whole scan ---
  v2f afrag[2][16];
#pragma unroll
  for (int t = 0; t < 2; ++t) {
    const int n = (tile0 + t) * 16 + mrow;            // pixel index
    const float* base = inp + (n >> 10) * VQ_IMGSTRIDE + (n & (VQ_HW - 1));
#pragma unroll
    for (int k = 0; k < 16; ++k) {
      const int c = 4 * k + khalf;
      v2f a;
      a.x = base[c * VQ_HW];
      a.y = base[(c + 1) * VQ_HW];
      afrag[t][k] = a;
    }
  }

  float minv[2][8];
  int   mini[2][8];
#pragma unroll
  for (int t = 0; t < 2; ++t)
#pragma unroll
    for (int r = 0; r < 8; ++r) { minv[t][r] = __builtin_inff(); mini[t][r] = 0; }

  // --- Scan the codebook, 16 codes per chunk ---
  for (int j0 = 0; j0 < VQ_NEMBED; j0 += 16) {
    const int col = j0 + (lane & 15);
    v8f acc0 = {};
    v8f acc1 = {};
#pragma unroll
    for (int k = 0; k < 16; ++k) {
      const int c = 4 * k + khalf;
      v2f b;
      b.x = emb[c * VQ_NEMBED + col];
      b.y = emb[(c + 1) * VQ_NEMBED + col];
      acc0 = __builtin_amdgcn_wmma_f32_16x16x4_f32(
          false, afrag[0][k], false, b, (short)0, acc0, false, false);
      acc1 = __builtin_amdgcn_wmma_f32_16x16x4_f32(
          false, afrag[1][k], false, b, (short)0, acc1, false, false);
    }
    const float e2c = e2[col];
#pragma unroll
    for (int r = 0; r < 8; ++r) {
      const float s0 = e2c - 2.0f * acc0[r];
      if (s0 < minv[0][r]) { minv[0][r] = s0; mini[0][r] = col; }
      const float s1 = e2c - 2.0f * acc1[r];
      if (s1 < minv[1][r]) { minv[1][r] = s1; mini[1][r] = col; }
    }
  }

  // --- Cross-lane reduction: row M=r lives in lanes 0..15 (VGPR r) and
  //     row M=r+8 in lanes 16..31.  Butterfly over the 16-lane half with
  //     lowest-index tie-break to match argmin semantics. ---
#pragma unroll
  for (int t = 0; t < 2; ++t) {
#pragma unroll
    for (int r = 0; r < 8; ++r) {
      float v  = minv[t][r];
      int   id = mini[t][r];
#pragma unroll
      for (int m = 1; m < 16; m <<= 1) {
        const float ov  = __shfl_xor(v, m, 32);
        const int   oid = __shfl_xor(id, m, 32);
        if (ov < v || (ov == v && oid < id)) { v = ov; id = oid; }
      }
      if ((lane & 15) == 0) {
        const int row = (tile0 + t) * 16 + r + (lane >> 4) * 8;
        ind[row] = id;
      }
    }
  }
}

// ---------------------------------------------------------------------------
// Kernel 3: gather codes -> quantize_st (transposed back to [B,C,H,W]),
// accumulate the MSE scalar, emit embed_ind as float.
// Grid: 64 blocks x 256 threads -> one thread per pixel; stores for a fixed c
// are coalesced across the wave (consecutive hw).
// ---------------------------------------------------------------------------
__global__ __launch_bounds__(256) void vq_gather_kernel(
    const float* __restrict__ inp, const float* __restrict__ emb,
    const int* __restrict__ ind, float* __restrict__ out_q,
    float* __restrict__ out_diff, float* __restrict__ out_ind) {
  const int n = blockIdx.x * 256 + threadIdx.x;   // 0..16383
  const int id = ind[n];
  out_ind[n] = (float)id;
  const int b  = n >> 10;
  const int hw = n & (VQ_HW - 1);
  const float* ibase = inp  + b * VQ_IMGSTRIDE + hw;
  float*       obase = out_q + b * VQ_IMGSTRIDE + hw;

  float acc = 0.0f;
#pragma unroll
  for (int c = 0; c < VQ_DIM; ++c) {
    const float q = emb[c * VQ_NEMBED + id];
    const float x = ibase[c * VQ_HW];
    const float d = q - x;
    acc += d * d;
    obase[c * VQ_HW] = q;      // quantize_st == quantize numerically
  }

  __shared__ float red[256];
  red[threadIdx.x] = acc;
  __syncthreads();
  for (int s = 128; s > 0; s >>= 1) {
    if (threadIdx.x < s) red[threadIdx.x] += red[threadIdx.x + s];
    __syncthreads();
  }
  if (threadIdx.x == 0)
    atomicAdd(out_diff, red[0] * (1.0f / (float)(VQ_NPIX * VQ_DIM)));
}

// ---------------------------------------------------------------------------
extern "C" void kernel_launch(void* const* d_in, const int* in_sizes, int n_in,
                              void* d_out, int out_size, void* d_ws, size_t ws_size,
                              hipStream_t stream) {
  const float* inp = (const float*)d_in[0];   // [16,64,32,32]
  const float* emb = (const float*)d_in[1];   // [64,8192]

  float* out      = (float*)d_out;
  float* out_q    = out;                        // 1,048,576 floats
  float* out_diff = out + (VQ_NPIX * VQ_DIM);   // 1 float
  float* out_ind  = out_diff + 1;               // 16,384 floats

  float* e2  = (float*)d_ws;                          // 8192 floats
  int*   ind = (int*)((char*)d_ws + VQ_NEMBED * 4);   // 16384 ints

  vq_prep_kernel<<<VQ_NEMBED / 256, 256, 0, stream>>>(emb, e2, out_diff);
  vq_argmin_kernel<<<64, 256, 0, stream>>>(inp, emb, e2, ind);
  vq_gather_kernel<<<VQ_NPIX / 256, 256, 0, stream>>>(inp, emb, ind, out_q,
                                                      out_diff, out_ind);
}